// BandSpecificFilter_20864951124432
// MI455X (gfx1250) — compile-verified
//
#include <hip/hip_runtime.h>
#include <cmath>
#include <complex>

#define T_LEN 5000
#define PAD   15
#define EXT_LEN (T_LEN + 2*PAD)   // 5030
#define NB    64
#define CIN   64
#define COUT  32
#define SEQ   (NB*COUT)           // 2048 sequences per band
#define TTILES 313                // ceil(5000/16)

typedef float v2f __attribute__((ext_vector_type(2)));
typedef float v8f __attribute__((ext_vector_type(8)));

// ---------------------------------------------------------------------------
// Kernel A: batched channel-mix GEMM via V_WMMA_F32_16X16X4_F32.
// Computes mix[band][t][b*32+o] = sum_c W[o][c] * x[b][c][t] for both bands.
// A fragment: 16(t) x 4(c) slice of x^T ; B fragment: 4(c) x 16(o) of W^T.
// D rows (N = o over lanes) store coalesced into the [t][seq] layout.
// ---------------------------------------------------------------------------
__global__ __launch_bounds__(256) void mix_gemm(
    const float* __restrict__ x,
    const float* __restrict__ Wd,
    const float* __restrict__ Wt,
    float* __restrict__ mix /* [2][T_LEN][SEQ] */) {
  // W repacked so that one lane's (K, K+1) pair is contiguous (ds_load_b64,
  // conflict-free across lanes): sW[(c>>1)*64 + o*2 + (c&1)]
  __shared__ float sWd[CIN * COUT];
  __shared__ float sWt[CIN * COUT];
  const int tid = threadIdx.x;
  for (int idx = tid; idx < CIN * COUT; idx += 256) {
    int o = idx >> 6, c = idx & 63;
    int dst = (c >> 1) * 64 + o * 2 + (c & 1);
    sWd[dst] = Wd[idx];
    sWt[dst] = Wt[idx];
  }
  __syncthreads();

  const int wave = blockIdx.x * 8 + (tid >> 5);   // 2504 blocks * 8 waves = 20032
  const int b  = wave / TTILES;
  const int tt = wave % TTILES;
  const int t0 = tt * 16;
  const int lane = tid & 31;
  const int half = lane >> 4;      // selects K=0/1 vs K=2/3 (A & B layouts)
  const int n    = lane & 15;      // M row (A) / N col (B)

  const float* xb = x + (size_t)b * CIN * T_LEN;
  int tc = t0 + n;                 // clamp tail loads; invalid D rows not stored
  if (tc > T_LEN - 1) tc = T_LEN - 1;

  v8f aD0 = {}, aD1 = {}, aT0 = {}, aT1 = {};
  for (int ks = 0; ks < 16; ++ks) {          // K = 64 channels, 4 per step
    const int c0 = 4 * ks + 2 * half;        // lane's first K element
    v2f af;
    af[0] = xb[(size_t)c0 * T_LEN + tc];
    af[1] = xb[(size_t)(c0 + 1) * T_LEN + tc];
    const int wb = (2 * ks + half) * 64 + n * 2;
    v2f bd0 = *reinterpret_cast<const v2f*>(&sWd[wb]);       // o-tile 0
    v2f bd1 = *reinterpret_cast<const v2f*>(&sWd[wb + 32]);  // o-tile 1
    v2f bt0 = *reinterpret_cast<const v2f*>(&sWt[wb]);
    v2f bt1 = *reinterpret_cast<const v2f*>(&sWt[wb + 32]);
    aD0 = __builtin_amdgcn_wmma_f32_16x16x4_f32(false, af, false, bd0, (short)0, aD0, false, false);
    aD1 = __builtin_amdgcn_wmma_f32_16x16x4_f32(false, af, false, bd1, (short)0, aD1, false, false);
    aT0 = __builtin_amdgcn_wmma_f32_16x16x4_f32(false, af, false, bt0, (short)0, aT0, false, false);
    aT1 = __builtin_amdgcn_wmma_f32_16x16x4_f32(false, af, false, bt1, (short)0, aT1, false, false);
  }

  // D layout: VGPR r holds row M = r (lanes 0-15) / M = r+8 (lanes 16-31), N = lane%16
  const size_t rowBase = (size_t)b * COUT;
  const size_t bandOff = (size_t)T_LEN * SEQ;
  for (int r = 0; r < 8; ++r) {
    const int t = t0 + r + 8 * half;
    if (t < T_LEN) {
      const size_t off = (size_t)t * SEQ + rowBase;
      mix[off + n]                 = aD0[r];
      mix[off + 16 + n]            = aD1[r];
      mix[bandOff + off + n]       = aT0[r];
      mix[bandOff + off + 16 + n]  = aT1[r];
    }
  }
}

// ---------------------------------------------------------------------------
// Kernel B: zero-phase forward-backward cascade of 2 biquads (DF-II-T).
// One wave per (band, batch); lane = output channel. All global streams are
// coalesced across the wave (stride SEQ floats per time step = 128 B/wave).
// Backward result overwrites the (dead) mix slice in place.
// ---------------------------------------------------------------------------
struct Biquad { float b0, b1, b2, a1, a2; };
struct BandCoef { Biquad s[2]; float zi[2][2]; };

__global__ __launch_bounds__(32) void filtfilt(
    float* __restrict__ mix, float* __restrict__ fwd,
    BandCoef cD, BandCoef cT) {
  const int band = blockIdx.x >> 6;
  const int b    = blockIdx.x & 63;
  const int lane = threadIdx.x;
  const BandCoef C = band ? cT : cD;

  float* m = mix + (size_t)band * T_LEN * SEQ + (size_t)b * COUT + lane;
  float* f = fwd + (size_t)band * EXT_LEN * SEQ + (size_t)b * COUT + lane;

  const float m0 = m[0];
  const float mL = m[(size_t)(T_LEN - 1) * SEQ];

  // ----- forward pass over odd-extended signal -----
  const float e0 = 2.f * m0 - m[(size_t)PAD * SEQ];   // ext[0]
  float z00 = C.zi[0][0] * e0, z01 = C.zi[0][1] * e0;
  float z10 = C.zi[1][0] * e0, z11 = C.zi[1][1] * e0;
  for (int i = 0; i < EXT_LEN; ++i) {
    float xt;
    if (i < PAD)                 xt = 2.f * m0 - m[(size_t)(PAD - i) * SEQ];
    else if (i < PAD + T_LEN)    xt = m[(size_t)(i - PAD) * SEQ];
    else                         xt = 2.f * mL - m[(size_t)(2 * (T_LEN - 1) + PAD - i) * SEQ];
    if (i + 17 < PAD + T_LEN && i >= PAD)
      __builtin_prefetch(&m[(size_t)(i + 17 - PAD) * SEQ], 0, 1);
    float y0 = C.s[0].b0 * xt + z00;
    z00 = C.s[0].b1 * xt - C.s[0].a1 * y0 + z01;
    z01 = C.s[0].b2 * xt - C.s[0].a2 * y0;
    float y1 = C.s[1].b0 * y0 + z10;
    z10 = C.s[1].b1 * y0 - C.s[1].a1 * y1 + z11;
    z11 = C.s[1].b2 * y0 - C.s[1].a2 * y1;
    f[(size_t)i * SEQ] = y1;
  }

  // ----- backward pass over reversed forward output -----
  const float eb = f[(size_t)(EXT_LEN - 1) * SEQ];
  z00 = C.zi[0][0] * eb; z01 = C.zi[0][1] * eb;
  z10 = C.zi[1][0] * eb; z11 = C.zi[1][1] * eb;
  for (int k = 0; k < EXT_LEN; ++k) {
    const float xt = f[(size_t)(EXT_LEN - 1 - k) * SEQ];
    if (k + 16 < EXT_LEN)
      __builtin_prefetch(&f[(size_t)(EXT_LEN - 17 - k) * SEQ], 0, 1);
    float y0 = C.s[0].b0 * xt + z00;
    z00 = C.s[0].b1 * xt - C.s[0].a1 * y0 + z01;
    z01 = C.s[0].b2 * xt - C.s[0].a2 * y0;
    float y1 = C.s[1].b0 * y0 + z10;
    z10 = C.s[1].b1 * y0 - C.s[1].a1 * y1 + z11;
    z11 = C.s[1].b2 * y0 - C.s[1].a2 * y1;
    const int t = (PAD + T_LEN - 1) - k;     // output sample index
    if ((unsigned)t < (unsigned)T_LEN)
      m[(size_t)t * SEQ] = y1;               // result overwrites mix slice
  }
}

// ---------------------------------------------------------------------------
// Kernel C: sum the two band results and transpose [t][b][o] -> [b][o][t]
// through LDS so both global streams are coalesced.
// ---------------------------------------------------------------------------
__global__ __launch_bounds__(256) void sum_transpose(
    const float* __restrict__ mix, float* __restrict__ out) {
  __shared__ float tile[32][33];
  const int blk = blockIdx.x;
  const int b   = blk / 157;
  const int tt  = blk % 157;
  const int t0  = tt * 32;
  const int tx  = threadIdx.x;   // 0..31
  const int ty  = threadIdx.y;   // 0..7
  const float* m0 = mix + (size_t)b * COUT;
  const float* m1 = mix + (size_t)T_LEN * SEQ + (size_t)b * COUT;
  for (int ph = 0; ph < 4; ++ph) {
    const int tr = ty + 8 * ph;
    const int t  = t0 + tr;
    float v = 0.f;
    if (t < T_LEN) v = m0[(size_t)t * SEQ + tx] + m1[(size_t)t * SEQ + tx];
    tile[tr][tx] = v;
  }
  __syncthreads();
  const int t = t0 + tx;
  if (t < T_LEN) {
    for (int ph = 0; ph < 4; ++ph) {
      const int o = ty + 8 * ph;
      out[((size_t)b * COUT + o) * T_LEN + t] = tile[tx][o];
    }
  }
}

// ---------------------------------------------------------------------------
// Host: Butterworth bandpass design (same zpk math as the reference) + zi.
// ---------------------------------------------------------------------------
static void design(double low, double high, BandCoef* bc) {
  const double PI = 3.14159265358979323846;
  const double fs = 20.0;
  std::complex<double> p[2] = {
    -std::exp(std::complex<double>(0.0, -PI / 4.0)),
    -std::exp(std::complex<double>(0.0,  PI / 4.0)) };
  const double wn0 = low / (fs / 2.0), wn1 = high / (fs / 2.0);
  const double w0 = 4.0 * std::tan(PI * wn0 / 2.0);
  const double w1 = 4.0 * std::tan(PI * wn1 / 2.0);
  const double bw = w1 - w0;
  const double wo = std::sqrt(w0 * w1);
  std::complex<double> pbp[4];
  for (int i = 0; i < 2; ++i) {
    std::complex<double> plp = p[i] * (bw / 2.0);
    std::complex<double> r = std::sqrt(plp * plp - wo * wo);
    pbp[i]     = plp + r;
    pbp[i + 2] = plp - r;
  }
  const double kbp = bw * bw;
  std::complex<double> pd[4], denom = 1.0;
  for (int i = 0; i < 4; ++i) {
    pd[i] = (4.0 + pbp[i]) / (4.0 - pbp[i]);
    denom *= (4.0 - pbp[i]);
  }
  const double kd = kbp * std::real(16.0 / denom);
  std::complex<double> pp[2]; int np = 0;
  for (int i = 0; i < 4; ++i) if (pd[i].imag() > 0.0) pp[np++] = pd[i];
  if (std::abs(pp[0] - 1.0) > std::abs(pp[1] - 1.0)) { std::complex<double> tmp = pp[0]; pp[0] = pp[1]; pp[1] = tmp; }

  double sos[2][6];
  sos[0][0] = kd;  sos[0][1] = -2.0 * kd; sos[0][2] = kd;
  sos[1][0] = 1.0; sos[1][1] =  2.0;      sos[1][2] = 1.0;
  for (int i = 0; i < 2; ++i) {
    sos[i][3] = 1.0;
    sos[i][4] = -2.0 * pp[i].real();
    sos[i][5] = std::norm(pp[i]);
  }
  double scale = 1.0;
  for (int s = 0; s < 2; ++s) {
    const double b0 = sos[s][0], b1 = sos[s][1], b2 = sos[s][2];
    const double a1 = sos[s][4], a2 = sos[s][5];
    const double det = (1.0 + a1) + a2;
    const double B0 = b1 - a1 * b0, B1 = b2 - a2 * b0;
    const double z0 = (B0 + B1) / det;
    const double z1 = (-a2 * B0 + (1.0 + a1) * B1) / det;
    bc->s[s].b0 = (float)b0; bc->s[s].b1 = (float)b1; bc->s[s].b2 = (float)b2;
    bc->s[s].a1 = (float)a1; bc->s[s].a2 = (float)a2;
    bc->zi[s][0] = (float)(scale * z0);
    bc->zi[s][1] = (float)(scale * z1);
    scale *= (b0 + b1 + b2) / (1.0 + a1 + a2);
  }
}

extern "C" void kernel_launch(void* const* d_in, const int* in_sizes, int n_in,
                              void* d_out, int out_size, void* d_ws, size_t ws_size,
                              hipStream_t stream) {
  (void)in_sizes; (void)n_in; (void)out_size; (void)ws_size;
  const float* x  = (const float*)d_in[0];
  const float* Wd = (const float*)d_in[1];
  const float* Wt = (const float*)d_in[2];
  float* out = (float*)d_out;

  float* mix = (float*)d_ws;                        // [2][T_LEN][SEQ]  (~82 MB)
  float* fwd = mix + (size_t)2 * T_LEN * SEQ;       // [2][EXT_LEN][SEQ] (~82 MB)

  BandCoef cD, cT;
  design(1.0, 4.0, &cD);
  design(4.0, 9.0, &cT);

  // 64 batches * 313 t-tiles = 20032 wave-tiles, 8 waves per 256-thread block
  mix_gemm<<<2504, 256, 0, stream>>>(x, Wd, Wt, mix);
  // 2 bands * 64 batches, one wave each
  filtfilt<<<128, 32, 0, stream>>>(mix, fwd, cD, cT);
  // 64 batches * 157 time-tiles of 32
  sum_transpose<<<dim3(64 * 157), dim3(32, 8), 0, stream>>>(mix, out);
}